// SGO_Loss_Assign_6751688589551
// MI455X (gfx1250) — compile-verified
//
#include <hip/hip_runtime.h>
#include <math.h>

// ---------------- configuration ----------------
#define E_MAX   768          // max edges per structure (expected ~540), multiple of 16
#define MAX_OPS 8
#define NMAX    64
#define BS      256
#define TN_TILES 4           // 16x16 tiles per wave strip (16x64 output per wave)

// ---------------- workspace layout (bytes) ----------------
// [0,64)      : int counts[16]   (counts[s], s=0 -> frac, s=1+k -> op k)
// [64,128)    : float sums[16]   (per-op assignment sums)
// [128, ...)  : per-structure edge data: E_MAX*3 floats ev + E_MAX floats |ev|^2
// [128K, ...) : per-op cost matrices, E_MAX x E_MAX f32, row pitch E_MAX
#define OFF_COUNTS   0
#define OFF_SUMS     64
#define OFF_EV       128
#define STRUCT_BYTES (E_MAX * 4 * 4)          // 3 floats ev + 1 float nrm per edge
#define OFF_COST     131072
#define COST_BYTES   ((size_t)E_MAX * E_MAX * 4)

typedef float v2f __attribute__((ext_vector_type(2)));
typedef float v8f __attribute__((ext_vector_type(8)));

__device__ __forceinline__ float* ev_ptr (char* ws, int s) { return (float*)(ws + OFF_EV + (size_t)s * STRUCT_BYTES); }
__device__ __forceinline__ float* nrm_ptr(char* ws, int s) { return ev_ptr(ws, s) + E_MAX * 3; }
__device__ __forceinline__ float* cost_ptr(char* ws, int k){ return (float*)(ws + OFF_COST + (size_t)k * COST_BYTES); }

// ================================================================
// Kernel 1: one block per structure (s==0: frac; s>=1: frac @ oprs[s-1].T mod 1)
// Enumerates periodic neighbor edge vectors (27 shifts, d <= 0.5) into ws.
// ================================================================
__global__ void build_edges(const float* __restrict__ frac,
                            const float* __restrict__ oprs,
                            int N, char* ws)
{
    __shared__ float fin[NMAX * 3];
    __shared__ float fs [NMAX * 3];
    __shared__ int   cnt;

    const int s   = blockIdx.x;
    const int tid = threadIdx.x;

    for (int t = tid; t < N * 3; t += BS) fin[t] = frac[t];
    if (tid == 0) cnt = 0;
    __syncthreads();

    if (s == 0) {
        for (int t = tid; t < N * 3; t += BS) fs[t] = fin[t];
    } else {
        const float* op = oprs + (size_t)(s - 1) * 9;
        for (int t = tid; t < N * 3; t += BS) {
            int a = t / 3, r = t - a * 3;
            float x = fin[a*3+0]*op[r*3+0] + fin[a*3+1]*op[r*3+1] + fin[a*3+2]*op[r*3+2];
            fs[t] = x - floorf(x);              // numpy % 1.0 semantics
        }
    }
    __syncthreads();

    float* ev  = ev_ptr (ws, s);
    float* nrm = nrm_ptr(ws, s);
    const float sv[3] = {0.0f, 1.0f, -1.0f};    // itertools.product([0,1,-1]) digit values

    const int total = N * 27 * N;
    for (int idx = tid; idx < total; idx += BS) {
        int src = idx / (27 * N);
        int rem = idx - src * 27 * N;
        int sh  = rem / N;
        int j   = rem - sh * N;
        float s0 = sv[sh / 9], s1 = sv[(sh / 3) % 3], s2 = sv[sh % 3];
        float ex = fs[j*3+0] + s0 - fs[src*3+0];
        float ey = fs[j*3+1] + s1 - fs[src*3+1];
        float ez = fs[j*3+2] + s2 - fs[src*3+2];
        float n2 = ex*ex + ey*ey + ez*ez;
        if (sqrtf(n2) <= 0.5f) {                // matches reference f32 norm compare
            int e = atomicAdd(&cnt, 1);         // LDS atomic (order irrelevant for LAP value)
            if (e < E_MAX) {
                ev[e*3+0] = ex; ev[e*3+1] = ey; ev[e*3+2] = ez;
                nrm[e] = n2;
            }
        }
    }
    __syncthreads();

    int c = cnt; if (c > E_MAX) c = E_MAX;
    int cpad = (c + 15) & ~15;                  // zero-pad to WMMA tile boundary
    for (int e = c + tid; e < cpad; e += BS) {
        ev[e*3+0] = 0.f; ev[e*3+1] = 0.f; ev[e*3+2] = 0.f; nrm[e] = 0.f;
    }
    if (tid == 0) ((int*)(ws + OFF_COUNTS))[s] = c;
}

// ================================================================
// Kernel 2: cost matrix via V_WMMA_F32_16X16X4_F32, 16x64 strip per wave.
// G = EV0 * EV1^T (K=3 padded to 4); cost = sqrt(max(|a|^2 + |b|^2 - 2G, 0)).
// A (16x4 f32) loaded once per wave: lanes 0-15 hold K0/K1, lanes 16-31 hold
// K2/K3 (K3 = 0 pad); B (4x16) mirrored by column; 4 WMMAs per wave.
// ================================================================
__global__ void cost_tiles(char* ws)
{
    const int tm     = blockIdx.x;
    const int tnBase = blockIdx.y * TN_TILES;
    const int k      = blockIdx.z;
    const int* counts = (const int*)(ws + OFF_COUNTS);
    const int E0 = counts[0], E1 = counts[1 + k];
    const int tilesM = (E0 + 15) >> 4, tilesN = (E1 + 15) >> 4;
    if (tm >= tilesM || tnBase >= tilesN) return;   // uniform -> EXEC all-1s below

    const float* ev0 = ev_ptr (ws, 0);
    const float* ev1 = ev_ptr (ws, 1 + k);
    const float* n0  = nrm_ptr(ws, 0);
    const float* n1  = nrm_ptr(ws, 1 + k);
    float* cost = cost_ptr(ws, k);

    const int lane = threadIdx.x;
    const int half = lane >> 4;                 // 0: K0/K1, 1: K2/K3
    const int l15  = lane & 15;

    // ---- A strip: loaded once, reused for TN_TILES WMMAs ----
    const float* arow = ev0 + 3 * (tm * 16 + l15);
    v2f A;                                      // branch-free lane split (keeps EXEC full)
    A.x = half ? arow[2] : arow[0];
    A.y = half ? 0.0f    : arow[1];

    float ni[8];                                // |ev0|^2 for this lane's 8 output rows
#pragma unroll
    for (int r = 0; r < 8; ++r) ni[r] = n0[tm * 16 + r + half * 8];

#pragma unroll
    for (int t = 0; t < TN_TILES; ++t) {
        const int tn = tnBase + t;
        if (tn >= tilesN) break;                // wave-uniform tail guard

        const float* bcol = ev1 + 3 * (tn * 16 + l15);
        v2f B;
        B.x = half ? bcol[2] : bcol[0];
        B.y = half ? 0.0f    : bcol[1];

        v8f C = {};
        C = __builtin_amdgcn_wmma_f32_16x16x4_f32(
                /*neg_a=*/false, A, /*neg_b=*/false, B,
                /*c_mod=*/(short)0, C, /*reuse_a=*/false, /*reuse_b=*/false);

        const int j  = tn * 16 + l15;
        const float nj = n1[j];
#pragma unroll
        for (int r = 0; r < 8; ++r) {           // C VGPR r: M = r (lanes 0-15) / r+8 (16-31)
            int i = tm * 16 + r + half * 8;
            float c2 = ni[r] + nj - 2.0f * C[r];
            cost[(size_t)i * E_MAX + j] = sqrtf(fmaxf(c2, 0.0f));
        }
    }
}

// ================================================================
// Kernel 3: Jonker-Volgenant shortest augmenting path LAP, one block per
// operator, column-parallel relax + argmin, f64 duals (matches scipy f64).
// Uses the identity assign_sum(C) == assign_sum(C^T): one solve per op.
// ================================================================
__global__ void lap_kernel(char* ws)
{
    const int k   = blockIdx.x;
    const int tid = threadIdx.x;
    const int* counts = (const int*)(ws + OFF_COUNTS);
    const int E0 = counts[0], E1 = counts[1 + k];
    const float* cost = cost_ptr(ws, k);
    const bool tr = E0 > E1;                    // _lsa transposes so nr <= nc
    const int nr = tr ? E1 : E0;
    const int nc = tr ? E0 : E1;

    __shared__ double u[E_MAX];
    __shared__ double v[E_MAX];
    __shared__ double shortest[E_MAX];
    __shared__ int    path[E_MAX];
    __shared__ int    row4col[E_MAX];
    __shared__ int    col4row[E_MAX];
    __shared__ unsigned char SC[E_MAX];
    __shared__ unsigned char SR[E_MAX];
    __shared__ double redv[BS];
    __shared__ int    redi[BS];
    __shared__ int    s_i, s_sink;
    __shared__ double s_minval;

    for (int t = tid; t < nc; t += BS) { v[t] = 0.0; row4col[t] = -1; }
    for (int t = tid; t < nr; t += BS) { u[t] = 0.0; col4row[t] = -1; }
    __syncthreads();

    for (int cur = 0; cur < nr; ++cur) {
        for (int t = tid; t < nc; t += BS) { shortest[t] = 1e300; SC[t] = 0; }
        for (int t = tid; t < nr; t += BS) SR[t] = 0;
        if (tid == 0) { s_i = cur; s_sink = -1; s_minval = 0.0; }
        __syncthreads();

        while (s_sink < 0) {
            const int    i    = s_i;
            const double minv = s_minval;
            const double ui   = u[i];
            if (tid == 0) SR[i] = 1;
            // relax all columns not yet in SC
            for (int j = tid; j < nc; j += BS) {
                if (!SC[j]) {
                    float cij = tr ? cost[(size_t)j * E_MAX + i]
                                   : cost[(size_t)i * E_MAX + j];
                    double d = minv + (double)cij - ui - v[j];
                    if (d < shortest[j]) { shortest[j] = d; path[j] = i; }
                }
            }
            __syncthreads();
            // block argmin over remaining columns
            double best = 1e300; int bj = E_MAX + 1;
            for (int j = tid; j < nc; j += BS)
                if (!SC[j] && shortest[j] < best) { best = shortest[j]; bj = j; }
            redv[tid] = best; redi[tid] = bj;
            __syncthreads();
            for (int sft = BS >> 1; sft > 0; sft >>= 1) {
                if (tid < sft) {
                    double ov = redv[tid + sft]; int oi = redi[tid + sft];
                    if (ov < redv[tid] || (ov == redv[tid] && oi < redi[tid])) {
                        redv[tid] = ov; redi[tid] = oi;
                    }
                }
                __syncthreads();
            }
            if (tid == 0) {
                int j = redi[0];
                s_minval = redv[0];
                SC[j] = 1;
                if (row4col[j] < 0) s_sink = j;
                else                s_i = row4col[j];
            }
            __syncthreads();
        }

        const double minv = s_minval;
        const int    sink = s_sink;
        // dual updates
        for (int r = tid; r < nr; r += BS) {
            if (SR[r]) {
                if (r == cur) u[r] += minv;
                else          u[r] += minv - shortest[col4row[r]];
            }
        }
        for (int j = tid; j < nc; j += BS)
            if (SC[j]) v[j] += shortest[j] - minv;
        __syncthreads();
        // augment along alternating path
        if (tid == 0) {
            int j = sink;
            while (true) {
                int i2 = path[j];
                row4col[j] = i2;
                int tmp = col4row[i2];
                col4row[i2] = j;
                j = tmp;
                if (i2 == cur) break;
            }
        }
        __syncthreads();
    }

    // sum assigned costs
    double acc = 0.0;
    for (int r = tid; r < nr; r += BS) {
        int j = col4row[r];
        float cij = tr ? cost[(size_t)j * E_MAX + r]
                       : cost[(size_t)r * E_MAX + j];
        acc += (double)cij;
    }
    redv[tid] = acc;
    __syncthreads();
    for (int sft = BS >> 1; sft > 0; sft >>= 1) {
        if (tid < sft) redv[tid] += redv[tid + sft];
        __syncthreads();
    }
    if (tid == 0) ((float*)(ws + OFF_SUMS))[k] = (float)redv[0];
}

// ================================================================
// Kernel 4: total = (sum_k 0.5*(s_k + s_k^T)) / n_ops = (sum_k s_k) / n_ops
// ================================================================
__global__ void finalize(char* ws, float* out, int n_ops)
{
    if (threadIdx.x == 0) {
        const float* sums = (const float*)(ws + OFF_SUMS);
        float t = 0.f;
        for (int k = 0; k < n_ops; ++k) t += sums[k];
        out[0] = t / (float)n_ops;
    }
}

// ================================================================
extern "C" void kernel_launch(void* const* d_in, const int* in_sizes, int n_in,
                              void* d_out, int out_size, void* d_ws, size_t ws_size,
                              hipStream_t stream)
{
    const float* frac = (const float*)d_in[0];   // [N,3] f32
    const float* oprs = (const float*)d_in[1];   // [n_ops,3,3] f32
    int N     = in_sizes[0] / 3;
    int n_ops = in_sizes[1] / 9;
    if (N > NMAX)        N = NMAX;
    if (n_ops > MAX_OPS) n_ops = MAX_OPS;
    if (n_ops < 1)       n_ops = 1;
    char* ws = (char*)d_ws;

    // 1) edge vectors for frac (block 0) and each transformed structure (blocks 1..n_ops)
    build_edges<<<n_ops + 1, BS, 0, stream>>>(frac, oprs, N, ws);

    // 2) cost matrices: one wave per 16x64 WMMA strip; excess strips exit early
    dim3 grid(E_MAX / 16, E_MAX / (16 * TN_TILES), n_ops);
    cost_tiles<<<grid, 32, 0, stream>>>(ws);

    // 3) one LAP per operator, operators concurrent across WGPs
    lap_kernel<<<n_ops, BS, 0, stream>>>(ws);

    // 4) reduce
    finalize<<<1, 32, 0, stream>>>(ws, (float*)d_out, n_ops);
}